// Combine_loss_87290915324262
// MI455X (gfx1250) — compile-verified
//
#include <hip/hip_runtime.h>
#include <hip/hip_bf16.h>

#define N_NODES 100000
// Padded to a multiple of 512 (8 waves x 64 floats/WMMA) so the WMMA
// reduction loop needs no bounds checks / EXEC divergence.
#define N_PAD   100352

typedef __attribute__((ext_vector_type(2))) float v2f;
typedef __attribute__((ext_vector_type(8))) float v8f;

// ---------------------------------------------------------------------------
// Kernel 1: zero the (padded) per-node accumulator in d_ws scratch.
// Padding lanes stay 0 forever -> contribute nothing to the reduction.
// ---------------------------------------------------------------------------
__global__ void zero_acc(float* __restrict__ acc, int n) {
    int i = blockIdx.x * blockDim.x + threadIdx.x;
    if (i < n) acc[i] = 0.0f;
}

// ---------------------------------------------------------------------------
// Kernel 2: scatter pass over 16M edges (the roofline-dominant work:
// 192 MB streamed once -> ~8us floor at 23.3 TB/s, plus 32M fp32 atomics
// that all land in the L2-resident 400KB acc region). Nontemporal loads
// keep the one-shot edge stream from evicting the hot acc lines.
// ---------------------------------------------------------------------------
__global__ void scatter_edges(const float* __restrict__ flow,
                              float* __restrict__ acc, int E) {
    int tid    = blockIdx.x * blockDim.x + threadIdx.x;
    int stride = gridDim.x * blockDim.x;
    for (int e = tid; e < E; e += stride) {
        const float* p = flow + 3ll * e;
        float s = __builtin_nontemporal_load(p + 0);
        float d = __builtin_nontemporal_load(p + 1);
        float v = __builtin_nontemporal_load(p + 2);
        int si = (int)s;
        int di = (int)d;
        atomicAdd(&acc[si],  v);
        atomicAdd(&acc[di], -v);
    }
}

// ---------------------------------------------------------------------------
// Kernel 3: full-precision fp32 reduction of acc[0..N_PAD) on the CDNA5
// matrix pipe: V_WMMA_F32_16X16X4_F32 with A = ones(16x4).
// D[m][n] = sum_k B[k][n] + C[m][n]  =>  sum over all 256 D elements =
// 16 * sum(the 64 B elements fed that iteration), independent of B's exact
// VGPR striping. One 256-thread block = 8 wave32 waves; branch-free inner
// loop: one global_load_b64 per lane + one WMMA per 64 floats.
// ---------------------------------------------------------------------------
__global__ void wmma_reduce(const float* __restrict__ acc,
                            float* __restrict__ out) {
    __shared__ float wave_sum[8];

    const int lane = threadIdx.x & 31;
    const int wave = threadIdx.x >> 5;

    v2f a;
    a.x = 1.0f;
    a.y = 1.0f;
    v8f c = {};

    const v2f* __restrict__ accv = (const v2f*)acc;

    // Each wave eats 64 floats (32 lanes x v2f) per WMMA; waves interleave.
    const int n_chunks = N_PAD / 64;          // 1568, multiple of 8
    int idx = wave * 32 + lane;               // v2f index
    for (int ch = wave; ch < n_chunks; ch += 8) {
        v2f b = accv[idx];
        // D = ones(16x4) * B(4x16) + C   (fp32, RNE accumulate)
        c = __builtin_amdgcn_wmma_f32_16x16x4_f32(
                /*neg_a=*/false, a, /*neg_b=*/false, b,
                /*c_mod=*/(short)0, c, /*reuse_a=*/false, /*reuse_b=*/false);
        idx += 8 * 32;
    }

    // Per-lane partial: sum this lane's 8 D registers.
    float local = c[0] + c[1] + c[2] + c[3] + c[4] + c[5] + c[6] + c[7];

    // Cross-lane reduce within wave32.
    for (int off = 16; off > 0; off >>= 1)
        local += __shfl_xor(local, off, 32);

    if (lane == 0) wave_sum[wave] = local;
    __syncthreads();

    if (threadIdx.x == 0) {
        float total = 0.0f;
        for (int w = 0; w < 8; ++w) total += wave_sum[w];
        // Every B element is counted once in each of the 16 D rows.
        out[0] = total * (1.0f / 16.0f);
    }
}

// ---------------------------------------------------------------------------
// Host-side launcher. Inputs: [pred, target, flow, c_flow]; only flow used.
// ---------------------------------------------------------------------------
extern "C" void kernel_launch(void* const* d_in, const int* in_sizes, int n_in,
                              void* d_out, int out_size, void* d_ws, size_t ws_size,
                              hipStream_t stream) {
    const float* flow = (const float*)d_in[2];
    const int E = in_sizes[2] / 3;

    float* acc = (float*)d_ws;       // N_PAD floats (~392 KB) of scratch
    float* out = (float*)d_out;      // single fp32 scalar

    // 1) clear accumulator (including padding)
    {
        int threads = 256;
        int blocks  = (N_PAD + threads - 1) / threads;
        zero_acc<<<blocks, threads, 0, stream>>>(acc, N_PAD);
    }

    // 2) scatter-add over all edges (grid-stride, enough waves to saturate)
    {
        int threads = 256;
        int blocks  = 4096;
        scatter_edges<<<blocks, threads, 0, stream>>>(flow, acc, E);
    }

    // 3) fp32 WMMA reduction to the final scalar
    {
        wmma_reduce<<<1, 256, 0, stream>>>(acc, out);
    }
}